// RungeKuttaIntegratorCell_cuda_9466107921020
// MI455X (gfx1250) — compile-verified
//
#include <hip/hip_runtime.h>
#include <hip/hip_bf16.h>

typedef float v2f __attribute__((ext_vector_type(2)));
typedef float v8f __attribute__((ext_vector_type(8)));

#if __has_builtin(__builtin_amdgcn_exp2f)
#define EXP2F(x) __builtin_amdgcn_exp2f(x)
#else
#define EXP2F(x) exp2f(x)
#endif

// RK4 integrator cell with RBF force term.
// Wave layout: each wave32 handles 16 elements. Lane L and L+16 both own
// element (L & 15). The lane half selects which pair of K-values of the
// 32-center RBF the lane evaluates per WMMA step, matching the documented
// f32 16x16x4 A/B VGPR layout (lanes 0-15 <-> K={0,1}, lanes 16-31 <-> K={2,3},
// VGPR0 <-> first K of pair, VGPR1 <-> second).
// A = w broadcast across all 16 rows, B = phi computed in-lane ->
// D[m][n] = f(elem n) for every m, so acc[0] in EVERY lane is f of its own
// element: the WMMA does the K-reduction and the broadcast with no shuffles.
__launch_bounds__(256)
__global__ void rk4_rbf_wmma_kernel(const float* __restrict__ u,
                                    const float* __restrict__ states,
                                    const float* __restrict__ centers,
                                    const float* __restrict__ log_sigmas,
                                    const float* __restrict__ rbf_w,
                                    const float* __restrict__ rbf_b,
                                    float* __restrict__ out,
                                    int n)
{
    constexpr float DT     = 0.005f;
    constexpr float M_INV  = 1.0f / 95.452f;
    constexpr float OFFST  = -3.2902f;
    constexpr float F_V    = 214.9261f;
    constexpr float F_C    = 19.3607f;
    constexpr float LOG2E  = 1.4426950408889634f;

    // kp[k] = { -(c_k * s_k), s_k, w_k, 0 }  with s_k = exp(log_sigma_k)
    __shared__ float4 kp[32];

    const int tid = threadIdx.x;
    if (tid < 32) {
        float c = centers[tid];
        float s = EXP2F(log_sigmas[tid] * LOG2E);   // e^ls
        kp[tid] = make_float4(-(c * s), s, rbf_w[tid], 0.0f);
    }
    __syncthreads();

    const int lane = tid & 31;
    const int wave = tid >> 5;
    const int koff = (lane >> 4) << 1;                       // 0 for lanes 0-15, 2 for 16-31
    const int base = blockIdx.x * 128 + wave * 16;           // first element of this wave
    if (base >= n) return;                                   // wave-uniform: EXEC stays full
    const int elem = base + (lane & 15);

    const float ub   = u[elem];
    const float y1   = states[2 * elem + 1];
    const float offb = OFFST + rbf_b[0];
    const float fvc  = F_V + F_C;

    // g(y) = (u - (F_V+F_C)*y - (OFFST + rbf(y))) * M_INV
    auto gfun = [&](float y) -> float {
        v8f acc = {0.0f, 0.0f, 0.0f, 0.0f, 0.0f, 0.0f, 0.0f, 0.0f};
#pragma unroll
        for (int t = 0; t < 8; ++t) {
            const int kk = 4 * t + koff;
            float4 p0 = kp[kk];
            float4 p1 = kp[kk + 1];
            // d = (y - c)*s = fma(y, s, -(c*s));  phi = exp(-d*d)
            float d0 = fmaf(y, p0.y, p0.x);
            float d1 = fmaf(y, p1.y, p1.x);
            float e0 = EXP2F(d0 * d0 * (-LOG2E));
            float e1 = EXP2F(d1 * d1 * (-LOG2E));
            v2f a = {p0.z, p1.z};   // A: every row = w[kk], w[kk+1]
            v2f b = {e0, e1};       // B: phi for this lane's element / K pair
            acc = __builtin_amdgcn_wmma_f32_16x16x4_f32(
                /*neg_a=*/false, a, /*neg_b=*/false, b,
                /*c_mod=*/(short)0, acc,
                /*reuse_a=*/false, /*reuse_b=*/false);
        }
        float f = acc[0];           // f(elem) replicated to every lane
        return (ub - fvc * y - offb - f) * M_INV;
    };

    const float h  = 0.5f * DT;
    const float c6 = DT / 6.0f;

    float g1 = gfun(y1);            // yd1 == k1_1
    float y2 = y1 + h * g1;         // k2_0 (== k3_0)
    float a2 = gfun(y2);            // k2_1 (== k3_1)
    float y4 = y1 + DT * g1;        // k4_0
    float a4 = gfun(y4);            // k4_1

    float st0 = c6 * (y1 + 2.0f * y2 + 2.0f * y2 + y4);
    float st1 = c6 * (g1 + 2.0f * a2 + 2.0f * a2 + a4);

    if (koff == 0) {                // lanes 0-15 own the store
        reinterpret_cast<float2*>(out)[elem] = make_float2(st0, st1);
    }
}

extern "C" void kernel_launch(void* const* d_in, const int* in_sizes, int n_in,
                              void* d_out, int out_size, void* d_ws, size_t ws_size,
                              hipStream_t stream) {
    const float* u        = (const float*)d_in[0];
    const float* states   = (const float*)d_in[1];
    const float* centers  = (const float*)d_in[2];
    const float* logsig   = (const float*)d_in[3];
    const float* w        = (const float*)d_in[4];
    const float* b        = (const float*)d_in[5];
    float*       out      = (float*)d_out;

    const int n = in_sizes[0];                 // 4,194,304
    const int elems_per_block = 128;           // 8 waves * 16 elements
    const int blocks = (n + elems_per_block - 1) / elems_per_block;

    rk4_rbf_wmma_kernel<<<blocks, 256, 0, stream>>>(u, states, centers, logsig, w, b, out, n);
}